// GAT_32744830665463
// MI455X (gfx1250) — compile-verified
//
#include <hip/hip_runtime.h>
#include <math.h>

#define NEG_SLOPE 0.2f

typedef __attribute__((ext_vector_type(2))) float v2f;
typedef __attribute__((ext_vector_type(8))) float v8f;

// ---------------------------------------------------------------------------
// float atomic max via sign-split int trick (init with -inf)
// ---------------------------------------------------------------------------
__device__ __forceinline__ void atomicMaxFloat(float* addr, float val) {
    if (val >= 0.0f)
        atomicMax((int*)addr, __float_as_int(val));
    else
        atomicMin((unsigned int*)addr, (unsigned int)__float_as_int(val));
}

// ---------------------------------------------------------------------------
// fill a float buffer
// ---------------------------------------------------------------------------
__global__ void gat_fill(float* __restrict__ p, float v, size_t n) {
    size_t i = (size_t)blockIdx.x * blockDim.x + threadIdx.x;
    if (i < n) p[i] = v;
}

// ---------------------------------------------------------------------------
// sum-reduce edge_attr (for self-loop fill = mean)
// ---------------------------------------------------------------------------
__global__ void gat_reduce_sum(const float* __restrict__ x, float* __restrict__ out, int n) {
    __shared__ float sm[256];
    float acc = 0.0f;
    for (int i = blockIdx.x * blockDim.x + threadIdx.x; i < n;
         i += gridDim.x * blockDim.x)
        acc += x[i];
    sm[threadIdx.x] = acc;
    __syncthreads();
    for (int s = 128; s > 0; s >>= 1) {
        if ((int)threadIdx.x < s) sm[threadIdx.x] += sm[threadIdx.x + s];
        __syncthreads();
    }
    if (threadIdx.x == 0) atomicAdd(out, sm[0]);
}

// ---------------------------------------------------------------------------
// Fused dual GEMM with fp32 WMMA:  Ya = X @ Wa, Yb = X @ Wb
// X:[nrows,K]  Wa,Wb:[K,NOUT]  Ya,Yb:[nrows,NOUT]   (NOUT = 16 or 32)
// One wave -> 16 rows x NOUT cols using V_WMMA_F32_16X16X4_F32.
// A layout (16x4 f32): lane l: M = l&15, VGPR v holds K = 2*(l>>4)+v
// B layout (4x16 f32): lane l: N = l&15, VGPR v holds K = 2*(l>>4)+v
// C/D layout (16x16 f32): VGPR v, lane l: M = v + 8*(l>>4), N = l&15
// ---------------------------------------------------------------------------
template <int K, int NOUT>
__global__ void gat_gemm_dual(const float* __restrict__ X,
                              const float* __restrict__ Wa,
                              const float* __restrict__ Wb,
                              float* __restrict__ Ya,
                              float* __restrict__ Yb,
                              int nrows) {
    constexpr int CT = NOUT / 16;  // column tiles
    int tid  = blockIdx.x * blockDim.x + threadIdx.x;
    int wave = tid >> 5;
    int lane = threadIdx.x & 31;
    int row0 = wave * 16;
    if (row0 >= nrows) return;  // wave-uniform exit: EXEC all-ones inside

    int half = lane >> 4;   // selects K pair {0,1} vs {2,3}
    int lr   = lane & 15;
    const float* xrow = X + (size_t)(row0 + lr) * K + half * 2;

    v8f zero = {0.f, 0.f, 0.f, 0.f, 0.f, 0.f, 0.f, 0.f};
    v8f accA[CT], accB[CT];
#pragma unroll
    for (int t = 0; t < CT; ++t) { accA[t] = zero; accB[t] = zero; }

    for (int k = 0; k < K; k += 4) {
        v2f a;
        a[0] = xrow[k];
        a[1] = xrow[k + 1];
        int krow = k + half * 2;
#pragma unroll
        for (int t = 0; t < CT; ++t) {
            v2f ba, bb;
            ba[0] = Wa[(size_t)krow * NOUT + t * 16 + lr];
            ba[1] = Wa[(size_t)(krow + 1) * NOUT + t * 16 + lr];
            bb[0] = Wb[(size_t)krow * NOUT + t * 16 + lr];
            bb[1] = Wb[(size_t)(krow + 1) * NOUT + t * 16 + lr];
            accA[t] = __builtin_amdgcn_wmma_f32_16x16x4_f32(
                false, a, false, ba, (short)0, accA[t], false, false);
            accB[t] = __builtin_amdgcn_wmma_f32_16x16x4_f32(
                false, a, false, bb, (short)0, accB[t], false, false);
        }
    }

#pragma unroll
    for (int t = 0; t < CT; ++t)
#pragma unroll
        for (int v = 0; v < 8; ++v) {
            int r = row0 + v + half * 8;
            Ya[(size_t)r * NOUT + t * 16 + lr] = accA[t][v];
            Yb[(size_t)r * NOUT + t * 16 + lr] = accB[t][v];
        }
}

// ---------------------------------------------------------------------------
// Edge pass 1: compute logit per edge (incl. self-loops), store, atomic-max
// H channels per edge, 32/H edges per wave. lane->channel for coalesced rows.
// ---------------------------------------------------------------------------
template <int H>
__global__ void gat_edge_logits(const float* __restrict__ xl,
                                const float* __restrict__ xr,
                                const float* __restrict__ We,
                                const float* __restrict__ att,
                                const int* __restrict__ srcs,
                                const int* __restrict__ dsts,
                                const float* __restrict__ eattr,
                                const float* __restrict__ sumbuf,
                                float* __restrict__ logits,
                                float* __restrict__ mbuf,
                                int nE, int nN) {
    constexpr int EPW = 32 / H;
    int tid  = blockIdx.x * blockDim.x + threadIdx.x;
    int wave = tid >> 5;
    int lane = threadIdx.x & 31;
    int sub  = lane / H;
    int ch   = lane % H;
    long long e = (long long)wave * EPW + sub;
    long long nTot = (long long)nE + nN;
    if (e >= nTot) return;

    int s, d;
    float ea;
    if (e < nE) {
        s  = srcs[e];
        d  = dsts[e];
        ea = eattr[e];
    } else {
        s = d = (int)(e - nE);
        ea = sumbuf[0] * (1.0f / (float)nE);  // mean fill for self-loops
    }
    float v = xl[(size_t)s * H + ch] + xr[(size_t)d * H + ch] + ea * We[ch];
    v = (v > 0.0f) ? v : v * NEG_SLOPE;   // leaky relu
    float t = v * att[ch];
#pragma unroll
    for (int m = H / 2; m >= 1; m >>= 1) t += __shfl_xor(t, m, 32);
    if (ch == 0) {
        logits[e] = t;
        atomicMaxFloat(&mbuf[d], t);
    }
}

// ---------------------------------------------------------------------------
// Edge pass 2: ex = exp(logit - m[dst]); accumulate denominator; store ex
// ---------------------------------------------------------------------------
__global__ void gat_edge_expsum(const int* __restrict__ dsts,
                                float* __restrict__ logits,
                                const float* __restrict__ mbuf,
                                float* __restrict__ den,
                                int nE, int nN) {
    long long e = (long long)blockIdx.x * blockDim.x + threadIdx.x;
    long long nTot = (long long)nE + nN;
    if (e >= nTot) return;
    int d = (e < nE) ? dsts[e] : (int)(e - nE);
    float ex = __expf(logits[e] - mbuf[d]);
    logits[e] = ex;  // reuse buffer for exp values
    atomicAdd(&den[d], ex);
}

// ---------------------------------------------------------------------------
// Edge pass 3: out[dst] += (ex/den[dst]) * xl[src]
// ---------------------------------------------------------------------------
template <int H>
__global__ void gat_edge_scatter(const float* __restrict__ xl,
                                 const int* __restrict__ srcs,
                                 const int* __restrict__ dsts,
                                 const float* __restrict__ exbuf,
                                 const float* __restrict__ den,
                                 float* __restrict__ out,
                                 int nE, int nN) {
    constexpr int EPW = 32 / H;
    int tid  = blockIdx.x * blockDim.x + threadIdx.x;
    int wave = tid >> 5;
    int lane = threadIdx.x & 31;
    int sub  = lane / H;
    int ch   = lane % H;
    long long e = (long long)wave * EPW + sub;
    long long nTot = (long long)nE + nN;
    if (e >= nTot) return;

    int s, d;
    if (e < nE) { s = srcs[e]; d = dsts[e]; }
    else        { s = d = (int)(e - nE); }
    float alpha = exbuf[e] / den[d];
    atomicAdd(&out[(size_t)d * H + ch], alpha * xl[(size_t)s * H + ch]);
}

// ---------------------------------------------------------------------------
// h = relu(h + b[c])   (in place, H power of two)
// ---------------------------------------------------------------------------
__global__ void gat_bias_relu(float* __restrict__ h, const float* __restrict__ b,
                              int hmask, size_t n) {
    size_t i = (size_t)blockIdx.x * blockDim.x + threadIdx.x;
    if (i >= n) return;
    float v = h[i] + b[i & hmask];
    h[i] = (v > 0.0f) ? v : 0.0f;
}

// ---------------------------------------------------------------------------
// scatter-mean pooling (accumulate sums + counts)
// ---------------------------------------------------------------------------
__global__ void gat_pool(const float* __restrict__ h, const int* __restrict__ batch,
                         float* __restrict__ pooled, float* __restrict__ cnt, int nN) {
    int i = blockIdx.x * blockDim.x + threadIdx.x;
    if (i >= nN * 16) return;
    int nd = i >> 4, c = i & 15;
    int g = batch[nd];
    atomicAdd(&pooled[g * 16 + c], h[i]);
    if (c == 0) atomicAdd(&cnt[g], 1.0f);
}

// ---------------------------------------------------------------------------
// out[g] = (pooled[g,:]/max(cnt,1)) @ W3 + b3
// ---------------------------------------------------------------------------
__global__ void gat_final(const float* __restrict__ pooled, const float* __restrict__ cnt,
                          const float* __restrict__ W3, const float* __restrict__ b3,
                          float* __restrict__ out, int G) {
    int g = blockIdx.x * blockDim.x + threadIdx.x;
    if (g >= G) return;
    float c = fmaxf(cnt[g], 1.0f);
    float acc = 0.0f;
#pragma unroll
    for (int i = 0; i < 16; ++i) acc += pooled[g * 16 + i] * W3[i];
    out[g] = acc / c + b3[0];
}

// ---------------------------------------------------------------------------
extern "C" void kernel_launch(void* const* d_in, const int* in_sizes, int n_in,
                              void* d_out, int out_size, void* d_ws, size_t ws_size,
                              hipStream_t stream) {
    const float* x     = (const float*)d_in[0];
    const int*   ei    = (const int*)d_in[1];   // [2,E] int32 (JAX x64 disabled)
    const float* eattr = (const float*)d_in[2]; // [E,1]
    const int*   batch = (const int*)d_in[3];   // [N]
    const float* Wl1 = (const float*)d_in[4];
    const float* Wr1 = (const float*)d_in[5];
    const float* We1 = (const float*)d_in[6];
    const float* att1 = (const float*)d_in[7];
    const float* b1  = (const float*)d_in[8];
    const float* Wl2 = (const float*)d_in[9];
    const float* Wr2 = (const float*)d_in[10];
    const float* We2 = (const float*)d_in[11];
    const float* att2 = (const float*)d_in[12];
    const float* b2  = (const float*)d_in[13];
    const float* W3  = (const float*)d_in[14];
    const float* b3  = (const float*)d_in[15];

    const int F  = 128, H1 = 32, H2 = 16;
    const int N  = in_sizes[0] / F;
    const int E  = in_sizes[2];
    const int G  = out_size;
    const long long nTot = (long long)E + N;

    const int* srcs = ei;
    const int* dsts = ei + E;

    // workspace carve-up
    float* w = (float*)d_ws;
    float* xl1    = w; w += (size_t)N * H1;
    float* xr1    = w; w += (size_t)N * H1;
    float* agg1   = w; w += (size_t)N * H1;   // attn out -> relu -> h
    float* xl2    = w; w += (size_t)N * H2;
    float* xr2    = w; w += (size_t)N * H2;
    float* agg2   = w; w += (size_t)N * H2;
    float* mbuf   = w; w += N;
    float* den    = w; w += N;
    float* logits = w; w += (size_t)nTot;
    float* pooled = w; w += (size_t)G * H2;
    float* cnt    = w; w += G;
    float* sumbuf = w; w += 1;

    auto nb = [](long long t) { return (int)((t + 255) / 256); };

    // edge_attr sum for self-loop mean fill
    gat_fill<<<1, 256, 0, stream>>>(sumbuf, 0.0f, 1);
    gat_reduce_sum<<<256, 256, 0, stream>>>(eattr, sumbuf, E);

    // ---- layer 1 projections: xl1 = x@Wl1, xr1 = x@Wr1 (WMMA f32) ----
    {
        long long waves = (N + 15) / 16;
        gat_gemm_dual<128, 32><<<nb(waves * 32), 256, 0, stream>>>(
            x, Wl1, Wr1, xl1, xr1, N);
    }

    // ---- layer 1 attention ----
    gat_fill<<<nb(N), 256, 0, stream>>>(mbuf, -INFINITY, (size_t)N);
    gat_fill<<<nb(N), 256, 0, stream>>>(den, 0.0f, (size_t)N);
    gat_fill<<<nb((long long)N * H1), 256, 0, stream>>>(agg1, 0.0f, (size_t)N * H1);

    gat_edge_logits<32><<<nb(nTot * 32), 256, 0, stream>>>(
        xl1, xr1, We1, att1, srcs, dsts, eattr, sumbuf, logits, mbuf, E, N);
    gat_edge_expsum<<<nb(nTot), 256, 0, stream>>>(dsts, logits, mbuf, den, E, N);
    gat_edge_scatter<32><<<nb(nTot * 32), 256, 0, stream>>>(
        xl1, srcs, dsts, logits, den, agg1, E, N);
    gat_bias_relu<<<nb((long long)N * H1), 256, 0, stream>>>(
        agg1, b1, H1 - 1, (size_t)N * H1);

    // ---- layer 2 projections: xl2 = h@Wl2, xr2 = h@Wr2 (WMMA f32) ----
    {
        long long waves = (N + 15) / 16;
        gat_gemm_dual<32, 16><<<nb(waves * 32), 256, 0, stream>>>(
            agg1, Wl2, Wr2, xl2, xr2, N);
    }

    // ---- layer 2 attention ----
    gat_fill<<<nb(N), 256, 0, stream>>>(mbuf, -INFINITY, (size_t)N);
    gat_fill<<<nb(N), 256, 0, stream>>>(den, 0.0f, (size_t)N);
    gat_fill<<<nb((long long)N * H2), 256, 0, stream>>>(agg2, 0.0f, (size_t)N * H2);

    gat_edge_logits<16><<<nb(((nTot + 1) / 2) * 32), 256, 0, stream>>>(
        xl2, xr2, We2, att2, srcs, dsts, eattr, sumbuf, logits, mbuf, E, N);
    gat_edge_expsum<<<nb(nTot), 256, 0, stream>>>(dsts, logits, mbuf, den, E, N);
    gat_edge_scatter<16><<<nb(((nTot + 1) / 2) * 32), 256, 0, stream>>>(
        xl2, srcs, dsts, logits, den, agg2, E, N);
    gat_bias_relu<<<nb((long long)N * H2), 256, 0, stream>>>(
        agg2, b2, H2 - 1, (size_t)N * H2);

    // ---- pooling + readout ----
    gat_fill<<<nb((long long)G * H2 + G), 256, 0, stream>>>(pooled, 0.0f, (size_t)G * H2 + G);
    gat_pool<<<nb((long long)N * 16), 256, 0, stream>>>(agg2, batch, pooled, cnt, N);
    gat_final<<<nb(G), 256, 0, stream>>>(pooled, cnt, W3, b3, (float*)d_out, G);
}